// MemStream_84482006712515
// MI455X (gfx1250) — compile-verified
//
#include <hip/hip_runtime.h>
#include <hip/hip_bf16.h>

#define T_DIM 2048
#define D_DIM 512
#define M_DIM 256
#define OUT_DIM 1024
#define BETA 800.0f
#define EPSV 1e-8f

#define BLK_N 512            // output columns per block
#define TILES_PER_WAVE 4     // 512 cols / 16 / 8 waves
#define A_PITCH 516          // 512 + 4 pad -> conflict-free column reads

typedef float v2f __attribute__((ext_vector_type(2)));
typedef float v8f __attribute__((ext_vector_type(8)));

// ---------------------------------------------------------------------------
// K1: fused normalize + encoder GEMM + bias + ReLU via V_WMMA_F32_16X16X4_F32.
// Block = 256 threads (8 waves), computes a 16x512 output strip.
// Phase 1: normalize the 16x512 x-tile ONCE into LDS (padded pitch 516).
// Phase 2: k-loop is pure data movement + WMMA: per k-step one ds_load_b64 for
//          the A fragment feeds 4 independent WMMAs (wave owns 4 N-tiles).
// ---------------------------------------------------------------------------
__global__ __launch_bounds__(256) void enc_gemm_kernel(
    const float* __restrict__ x, const float* __restrict__ W,
    const float* __restrict__ bias, const float* __restrict__ mean,
    const float* __restrict__ stdv, float* __restrict__ enc) {
  __shared__ float sA[16][A_PITCH];  // normalized x tile
  __shared__ float smean[D_DIM];
  __shared__ float sinv[D_DIM];      // 0 for degenerate columns, else 1/std
  const int tid = threadIdx.x;
  for (int i = tid; i < D_DIM; i += 256) {
    const float sd = stdv[i];
    smean[i] = mean[i];
    sinv[i] = (sd < EPSV) ? 0.0f : (1.0f / sd);
  }
  __syncthreads();

  const int tileM = blockIdx.x;  // 0..127
  {
    // 16 rows x 512 cols = 8192 elems; each thread normalizes 32 contiguous.
    const int r = tid >> 4;
    const int c0 = (tid & 15) * 32;
    const float* xr = x + (size_t)(tileM * 16 + r) * D_DIM + c0;
#pragma unroll 8
    for (int j = 0; j < 32; ++j)
      sA[r][c0 + j] = (xr[j] - smean[c0 + j]) * sinv[c0 + j];
  }
  __syncthreads();

  const int lane = tid & 31;
  const int wave = tid >> 5;
  const int col = blockIdx.y * BLK_N + wave * (TILES_PER_WAVE * 16) + (lane & 15);
  const int arow = lane & 15;
  const int khalf = (lane < 16) ? 0 : 2;

  v8f acc[TILES_PER_WAVE];
#pragma unroll
  for (int nt = 0; nt < TILES_PER_WAVE; ++nt) acc[nt] = (v8f){};

#pragma unroll 4
  for (int k = 0; k < D_DIM; k += 4) {
    const int kk = k + khalf;
    v2f a;
    a.x = sA[arow][kk];
    a.y = sA[arow][kk + 1];
    const float* w0 = W + (size_t)kk * OUT_DIM + col;
    const float* w1 = w0 + OUT_DIM;
#pragma unroll
    for (int nt = 0; nt < TILES_PER_WAVE; ++nt) {
      v2f b;
      b.x = w0[nt * 16];
      b.y = w1[nt * 16];
      acc[nt] = __builtin_amdgcn_wmma_f32_16x16x4_f32(
          /*neg_a=*/false, a, /*neg_b=*/false, b,
          /*c_mod=*/(short)0, acc[nt], /*reuse_a=*/false, /*reuse_b=*/false);
    }
  }

  const int mOff = (lane >= 16) ? 8 : 0;
#pragma unroll
  for (int nt = 0; nt < TILES_PER_WAVE; ++nt) {
    const int n = col + nt * 16;
    const float bn = bias[n];
#pragma unroll
    for (int j = 0; j < 8; ++j) {
      const int m = tileM * 16 + mOff + j;
      const float v = acc[nt][j] + bn;
      enc[(size_t)m * OUT_DIM + n] = fmaxf(v, 0.0f);
    }
  }
}

// ---------------------------------------------------------------------------
// K2: L1 cdist + top-3 mean score. One block per stream sample t.
// ---------------------------------------------------------------------------
__global__ __launch_bounds__(256) void knn_score_kernel(
    const float* __restrict__ enc, const float* __restrict__ memory,
    float* __restrict__ scores) {
  __shared__ float erow[OUT_DIM];
  __shared__ float dist[M_DIM];
  const int t = blockIdx.x;
  const int tid = threadIdx.x;
  for (int i = tid; i < OUT_DIM; i += 256) erow[i] = enc[(size_t)t * OUT_DIM + i];
  __syncthreads();

  const int lane = tid & 31;
  const int wave = tid >> 5;
  for (int pass = 0; pass < M_DIM / 8; ++pass) {
    const int m = pass * 8 + wave;
    const float* mrow = memory + (size_t)m * OUT_DIM;
    if (pass + 1 < M_DIM / 8)  // gfx1250 global_prefetch_b8 of the next row
      __builtin_prefetch(mrow + 8 * OUT_DIM + lane * 32, 0, 1);
    float s = 0.0f;
#pragma unroll
    for (int d = lane; d < OUT_DIM; d += 32) s += fabsf(erow[d] - mrow[d]);
#pragma unroll
    for (int off = 16; off > 0; off >>= 1) s += __shfl_xor(s, off, 32);
    if (lane == 0) dist[m] = s;
  }
  __syncthreads();

  if (tid == 0) {
    float d0 = 1e30f, d1 = 1e30f, d2 = 1e30f;
    for (int m = 0; m < M_DIM; ++m) {
      const float v = dist[m];
      if (v < d0)      { d2 = d1; d1 = d0; d0 = v; }
      else if (v < d1) { d2 = d1; d1 = v; }
      else if (v < d2) { d2 = v; }
    }
    scores[t] = (d0 + d1 + d2) * (1.0f / 3.0f);
  }
}

// ---------------------------------------------------------------------------
// K3: sequential-semantics threshold-gated circular scatter update.
// ---------------------------------------------------------------------------
__global__ __launch_bounds__(256) void update_kernel(
    const float* __restrict__ scores, const float* __restrict__ x,
    const float* __restrict__ memory, const float* __restrict__ mem_data,
    const float* __restrict__ enc, const int* __restrict__ countp,
    float* __restrict__ newMemory, float* __restrict__ newMemData,
    int* __restrict__ nAcceptOut) {
  __shared__ int chunkCnt[256];
  __shared__ int chunkOff[256];
  __shared__ int acceptedIdx[T_DIM];
  __shared__ int writer[M_DIM];
  __shared__ int sNA;
  const int tid = threadIdx.x;
  const int base = tid * 8;

  int cnt = 0;
#pragma unroll
  for (int j = 0; j < 8; ++j) cnt += (scores[base + j] <= BETA) ? 1 : 0;
  chunkCnt[tid] = cnt;
  __syncthreads();
  if (tid == 0) {
    int run = 0;
    for (int i = 0; i < 256; ++i) { chunkOff[i] = run; run += chunkCnt[i]; }
    sNA = run;
    *nAcceptOut = run;
  }
  __syncthreads();

  int r = chunkOff[tid];
#pragma unroll
  for (int j = 0; j < 8; ++j) {
    const int t = base + j;
    if (scores[t] <= BETA) acceptedIdx[r++] = t;
  }
  __syncthreads();

  const int nAccept = sNA;
  const int count = *countp;
  {
    const int s = tid;  // slot 0..255
    const int r0 = ((s - count) % M_DIM + M_DIM) % M_DIM;
    int wtr = -1;
    if (r0 < nAccept) {
      const int rLast = r0 + ((nAccept - 1 - r0) / M_DIM) * M_DIM;
      wtr = acceptedIdx[rLast];
    }
    writer[s] = wtr;
  }
  __syncthreads();

  for (int idx = tid; idx < M_DIM * OUT_DIM; idx += 256) {
    const int s = idx >> 10, d = idx & (OUT_DIM - 1);
    const int wtr = writer[s];
    newMemory[idx] = (wtr >= 0) ? enc[(size_t)wtr * OUT_DIM + d] : memory[idx];
  }
  for (int idx = tid; idx < M_DIM * D_DIM; idx += 256) {
    const int s = idx >> 9, d = idx & (D_DIM - 1);
    const int wtr = writer[s];
    newMemData[idx] = (wtr >= 0) ? x[(size_t)wtr * D_DIM + d] : mem_data[idx];
  }
}

// ---------------------------------------------------------------------------
// K4: column mean + unbiased std over updated mem_data (gate on new_count>0).
// ---------------------------------------------------------------------------
__global__ __launch_bounds__(512) void stats_kernel(
    const float* __restrict__ newMemData, const float* __restrict__ oldMean,
    const float* __restrict__ oldStd, const int* __restrict__ countp,
    const int* __restrict__ nAcceptp, float* __restrict__ newMean,
    float* __restrict__ newStd) {
  const int c = threadIdx.x;  // column 0..511
  float sum = 0.0f, sumsq = 0.0f;
  for (int rr = 0; rr < M_DIM; ++rr) {
    const float v = newMemData[(size_t)rr * D_DIM + c];
    sum += v;
    sumsq += v * v;
  }
  const float mu = sum * (1.0f / M_DIM);
  float var = (sumsq - (float)M_DIM * mu * mu) * (1.0f / (M_DIM - 1));
  var = fmaxf(var, 0.0f);
  float sd = sqrtf(var);
  if (sd < EPSV) sd = 1.0f;  // EPS clamp BEFORE the count gate (matches reference)
  const int newCount = *countp + *nAcceptp;
  newMean[c] = (newCount > 0) ? mu : oldMean[c];
  newStd[c]  = (newCount > 0) ? sd : oldStd[c];
}

extern "C" void kernel_launch(void* const* d_in, const int* in_sizes, int n_in,
                              void* d_out, int out_size, void* d_ws, size_t ws_size,
                              hipStream_t stream) {
  (void)in_sizes; (void)n_in; (void)out_size; (void)ws_size;
  const float* x        = (const float*)d_in[0];  // [2048,512]
  const float* memory   = (const float*)d_in[1];  // [256,1024]
  const float* mem_data = (const float*)d_in[2];  // [256,512]
  const float* W_enc    = (const float*)d_in[3];  // [512,1024]
  const float* b_enc    = (const float*)d_in[4];  // [1024]
  const float* mean     = (const float*)d_in[5];  // [512]
  const float* stdv     = (const float*)d_in[6];  // [512]
  const int*   count    = (const int*)d_in[7];    // scalar

  float* out = (float*)d_out;
  float* scores     = out;                                   // [2048]
  float* newMemory  = scores + T_DIM;                        // [256*1024]
  float* newMemData = newMemory + (size_t)M_DIM * OUT_DIM;   // [256*512]
  float* newMean    = newMemData + (size_t)M_DIM * D_DIM;    // [512]
  float* newStd     = newMean + D_DIM;                       // [512]

  float* enc = (float*)d_ws;                                 // [2048*1024] f32
  int* nAcceptOut = (int*)((char*)d_ws + (size_t)T_DIM * OUT_DIM * sizeof(float));

  enc_gemm_kernel<<<dim3(T_DIM / 16, OUT_DIM / BLK_N), 256, 0, stream>>>(
      x, W_enc, b_enc, mean, stdv, enc);
  knn_score_kernel<<<T_DIM, 256, 0, stream>>>(enc, memory, scores);
  update_kernel<<<1, 256, 0, stream>>>(scores, x, memory, mem_data, enc, count,
                                       newMemory, newMemData, nAcceptOut);
  stats_kernel<<<1, 512, 0, stream>>>(newMemData, mean, stdv, count, nAcceptOut,
                                      newMean, newStd);
}